// GATAutoEncoder_20710332301465
// MI455X (gfx1250) — compile-verified
//
#include <hip/hip_runtime.h>
#include <hip/hip_bf16.h>

// ---------------------------------------------------------------------------
// GATv2 autoencoder forward for MI455X (gfx1250), wave32, WMMA f32 16x16x4.
// Register-blocked WMMA GEMM: each wave owns a 32 x (16*NT) tile (MT=2),
// so every B fragment feeds two v_wmma_f32_16x16x4_f32 ops.
// ---------------------------------------------------------------------------

typedef float v2f __attribute__((ext_vector_type(2)));
typedef float v8f __attribute__((ext_vector_type(8)));

#define NNODES 50000
#define NEDGES 300000
#define DIN    1024
#define DHID   512
#define DOUT   30
#define DOUTP  32
#define NEG_SLOPE 0.2f

// ---------------------------------------------------------------------------
// C[M,N] = A[M,K] @ B[K,N], row-major. N % (16*NT) == 0, K % 4 == 0,
// M % 16 == 0 (M % 32 may be 16 -> handled by `has2`).
//
// Fragment layouts per CDNA5 ISA 7.12.2:
//   A 16x4 : lanes 0-15 rows M=0..15 {K=0,1}; lanes 16-31 rows M=0..15 {K=2,3}
//   B 4x16 : lanes 0-15 cols N=0..15 {K=0,1}; lanes 16-31 cols N=0..15 {K=2,3}
//   C 16x16: lanes 0-15 cols N=0..15, VGPR r = row M=r; lanes 16-31 row M=8+r
// ---------------------------------------------------------------------------
template <int NT>
__global__ void wmma_gemm_f32(const float* __restrict__ A,
                              const float* __restrict__ B,
                              float* __restrict__ C,
                              int M, int N, int K) {
    const int wave_id = (blockIdx.x * blockDim.x + threadIdx.x) >> 5;
    const int lane    = threadIdx.x & 31;
    const int tilesN  = N / (16 * NT);
    const int m0      = (wave_id / tilesN) * 32;       // 32-row supertile
    const int n0      = (wave_id % tilesN) * (16 * NT);
    if (m0 >= M) return;
    const bool has2 = (m0 + 16) < M;

    const int half = lane >> 4;   // 0: K lo pair, 1: K hi pair
    const int l15  = lane & 15;

    v8f acc0[NT], acc1[NT];
    const v8f zero = {0.f, 0.f, 0.f, 0.f, 0.f, 0.f, 0.f, 0.f};
#pragma unroll
    for (int t = 0; t < NT; ++t) { acc0[t] = zero; acc1[t] = zero; }

    const float* arow0 = A + (size_t)(m0 + l15) * K;
    const float* arow1 = A + (size_t)(m0 + 16 + l15) * K;

    if (has2) {
        for (int k = 0; k < K; k += 4) {
            const int ka = k + half * 2;
            v2f a0 = *(const v2f*)(arow0 + ka);        // global_load_b64
            v2f a1 = *(const v2f*)(arow1 + ka);
#pragma unroll
            for (int t = 0; t < NT; ++t) {
                const int bn = n0 + t * 16 + l15;
                v2f b;
                b.x = B[(size_t)(ka + 0) * N + bn];
                b.y = B[(size_t)(ka + 1) * N + bn];
                acc0[t] = __builtin_amdgcn_wmma_f32_16x16x4_f32(
                    false, a0, false, b, (short)0, acc0[t], false, false);
                acc1[t] = __builtin_amdgcn_wmma_f32_16x16x4_f32(
                    false, a1, false, b, (short)0, acc1[t], false, false);
            }
        }
    } else {
        for (int k = 0; k < K; k += 4) {
            const int ka = k + half * 2;
            v2f a0 = *(const v2f*)(arow0 + ka);
#pragma unroll
            for (int t = 0; t < NT; ++t) {
                const int bn = n0 + t * 16 + l15;
                v2f b;
                b.x = B[(size_t)(ka + 0) * N + bn];
                b.y = B[(size_t)(ka + 1) * N + bn];
                acc0[t] = __builtin_amdgcn_wmma_f32_16x16x4_f32(
                    false, a0, false, b, (short)0, acc0[t], false, false);
            }
        }
    }

#pragma unroll
    for (int t = 0; t < NT; ++t) {
        const int col = n0 + t * 16 + l15;
#pragma unroll
        for (int r = 0; r < 8; ++r) {
            C[(size_t)(m0 + half * 8 + r) * N + col] = acc0[t][r];
        }
    }
    if (has2) {
#pragma unroll
        for (int t = 0; t < NT; ++t) {
            const int col = n0 + t * 16 + l15;
#pragma unroll
            for (int r = 0; r < 8; ++r) {
                C[(size_t)(m0 + 16 + half * 8 + r) * N + col] = acc1[t][r];
            }
        }
    }
}

// ---------------------------------------------------------------------------
// Edge score: e_j = att . leaky_relu(h[src]+h[dst]); segment max via
// order-preserving uint atomicMax. One wave32 per edge; h is L2-resident.
// ---------------------------------------------------------------------------
__device__ __forceinline__ unsigned f32_encode(float f) {
    unsigned u = __float_as_uint(f);
    return (u & 0x80000000u) ? ~u : (u | 0x80000000u);
}
__device__ __forceinline__ float f32_decode(unsigned u) {
    return __uint_as_float((u & 0x80000000u) ? (u & 0x7FFFFFFFu) : ~u);
}

__global__ void edge_scores(const float* __restrict__ h,
                            const float* __restrict__ att,
                            const int* __restrict__ src,
                            const int* __restrict__ dst,
                            float* __restrict__ e,
                            unsigned* __restrict__ mEnc,
                            int E) {
    const int wid  = (blockIdx.x * blockDim.x + threadIdx.x) >> 5;
    const int lane = threadIdx.x & 31;
    if (wid >= E) return;
    const int s = src[wid], d = dst[wid];
    const float* hs = h + (size_t)s * DHID;
    const float* hd = h + (size_t)d * DHID;
    float a = 0.f;
#pragma unroll
    for (int i = lane; i < DHID; i += 32) {
        float v = hs[i] + hd[i];
        v = v > 0.f ? v : v * NEG_SLOPE;
        a += att[i] * v;
    }
#pragma unroll
    for (int off = 16; off > 0; off >>= 1) a += __shfl_xor(a, off, 32);
    if (lane == 0) {
        e[wid] = a;
        atomicMax(mEnc + d, f32_encode(a));
    }
}

// ex_j = exp(e_j - m[dst_j]); den[dst_j] += ex_j  (in-place over e)
__global__ void edge_exp(float* __restrict__ e,
                         const int* __restrict__ dst,
                         const unsigned* __restrict__ mEnc,
                         float* __restrict__ den, int E) {
    const int j = blockIdx.x * blockDim.x + threadIdx.x;
    if (j >= E) return;
    const int d = dst[j];
    const float m = f32_decode(mEnc[d]);
    const float v = expf(e[j] - m);
    e[j] = v;
    atomicAdd(den + d, v);
}

// alpha_j = ex_j / den[dst_j]  (in-place)
__global__ void edge_norm(float* __restrict__ ex,
                          const int* __restrict__ dst,
                          const float* __restrict__ den, int E) {
    const int j = blockIdx.x * blockDim.x + threadIdx.x;
    if (j >= E) return;
    ex[j] = ex[j] / den[dst[j]];
}

// acc[dst] += alpha * x[src]   (wave per edge; f32 atomics land in L2)
__global__ void edge_aggregate(const float* __restrict__ x,
                               const float* __restrict__ alpha,
                               const int* __restrict__ src,
                               const int* __restrict__ dst,
                               float* __restrict__ acc, int E) {
    const int wid  = (blockIdx.x * blockDim.x + threadIdx.x) >> 5;
    const int lane = threadIdx.x & 31;
    if (wid >= E) return;
    const int s = src[wid], d = dst[wid];
    const float a = alpha[wid];
    const float* xs = x + (size_t)s * DHID;
    float* out = acc + (size_t)d * DHID;
#pragma unroll
    for (int i = lane; i < DHID; i += 32) atomicAdd(out + i, a * xs[i]);
}

// elementwise ELU in place
__global__ void elu_inplace(float* __restrict__ x, int n) {
    const int i = blockIdx.x * blockDim.x + threadIdx.x;
    if (i >= n) return;
    const float v = x[i];
    x[i] = v > 0.f ? v : expm1f(v);
}

// W2pad[k,0:32] = [W2[k,0:30], 0, 0]; W2padT = transpose(W2pad)
__global__ void pad_w2(const float* __restrict__ W2,
                       float* __restrict__ W2pad,
                       float* __restrict__ W2padT) {
    const int idx = blockIdx.x * blockDim.x + threadIdx.x;
    if (idx >= DHID * DOUTP) return;
    const int k = idx >> 5, c = idx & 31;
    const float v = (c < DOUT) ? W2[k * DOUT + c] : 0.f;
    W2pad[k * DOUTP + c] = v;
    W2padT[c * DHID + k] = v;
}

// out[N,30] = h2pad[N,32][:, :30]
__global__ void slice_h2(const float* __restrict__ h2pad,
                         float* __restrict__ out) {
    const int idx = blockIdx.x * blockDim.x + threadIdx.x;
    if (idx >= NNODES * DOUT) return;
    const int r = idx / DOUT, c = idx - r * DOUT;
    out[idx] = h2pad[r * DOUTP + c];
}

// ---------------------------------------------------------------------------
extern "C" void kernel_launch(void* const* d_in, const int* in_sizes, int n_in,
                              void* d_out, int out_size, void* d_ws, size_t ws_size,
                              hipStream_t stream) {
    const float* features = (const float*)d_in[0];
    const int*   eidx     = (const int*)d_in[1];
    const float* W1       = (const float*)d_in[2];
    const float* att1     = (const float*)d_in[3];
    const float* W2       = (const float*)d_in[4];
    const float* W4       = (const float*)d_in[5];
    float*       out      = (float*)d_out;

    const int* src = eidx;            // edge_index[0, :]
    const int* dst = eidx + NEDGES;   // edge_index[1, :]

    // ---- workspace carve-up --------------------------------------------
    char* ws = (char*)d_ws;
    float*    h      = (float*)ws;                 ws += (size_t)NNODES * DHID * 4;  // h, later g
    float*    acc    = (float*)ws;                 ws += (size_t)NNODES * DHID * 4;  // h1 acc, later h3 acc
    float*    h2pad  = (float*)ws;                 ws += (size_t)NNODES * DOUTP * 4;
    float*    ealpha = (float*)ws;                 ws += (size_t)NEDGES * 4;         // e -> ex -> alpha
    unsigned* mEnc   = (unsigned*)ws;              ws += (size_t)NNODES * 4;
    float*    den    = (float*)ws;                 ws += (size_t)NNODES * 4;
    float*    W2pad  = (float*)ws;                 ws += (size_t)DHID * DOUTP * 4;
    float*    W2padT = (float*)ws;                 ws += (size_t)DHID * DOUTP * 4;

    const int superM = (NNODES + 31) / 32;         // 1563 32-row supertiles

    // ---- init ----------------------------------------------------------
    hipMemsetAsync(mEnc, 0, (size_t)NNODES * 4, stream);        // encoded -inf == 0
    hipMemsetAsync(den,  0, (size_t)NNODES * 4, stream);
    hipMemsetAsync(acc,  0, (size_t)NNODES * DHID * 4, stream);
    pad_w2<<<(DHID * DOUTP + 255) / 256, 256, 0, stream>>>(W2, W2pad, W2padT);

    // ---- conv1: h = X @ W1   [50000,1024]@[1024,512] -------------------
    {   const int waves = superM * (DHID / 64);    // NT=4 -> 32x64 tile
        wmma_gemm_f32<4><<<(waves + 7) / 8, 256, 0, stream>>>(features, W1, h,
                                                              NNODES, DHID, DIN);
    }

    // ---- GATv2 attention ----------------------------------------------
    edge_scores<<<(NEDGES + 7) / 8, 256, 0, stream>>>(h, att1, src, dst,
                                                      ealpha, mEnc, NEDGES);
    edge_exp<<<(NEDGES + 255) / 256, 256, 0, stream>>>(ealpha, dst, mEnc, den, NEDGES);
    edge_norm<<<(NEDGES + 255) / 256, 256, 0, stream>>>(ealpha, dst, den, NEDGES);

    // ---- h1 = elu(aggregate(h, alpha)) ---------------------------------
    edge_aggregate<<<(NEDGES + 7) / 8, 256, 0, stream>>>(h, ealpha, src, dst,
                                                         acc, NEDGES);
    elu_inplace<<<(NNODES * DHID + 255) / 256, 256, 0, stream>>>(acc, NNODES * DHID);

    // ---- conv2: h2 = h1 @ W2   (padded to 32 cols) ---------------------
    {   const int waves = superM;                  // NT=2 -> 32x32 tile
        wmma_gemm_f32<2><<<(waves + 7) / 8, 256, 0, stream>>>(acc, W2pad, h2pad,
                                                              NNODES, DOUTP, DHID);
    }
    slice_h2<<<(NNODES * DOUT + 255) / 256, 256, 0, stream>>>(h2pad, out);

    // ---- conv3: g = h2pad @ W2padT  [50000,32]@[32,512] ----------------
    {   const int waves = superM * (DHID / 64);
        wmma_gemm_f32<4><<<(waves + 7) / 8, 256, 0, stream>>>(h2pad, W2padT, h,
                                                              NNODES, DHID, DOUTP);
    }

    // ---- h3 = elu(aggregate(g, alpha)) ---------------------------------
    hipMemsetAsync(acc, 0, (size_t)NNODES * DHID * 4, stream);   // reuse acc
    edge_aggregate<<<(NEDGES + 7) / 8, 256, 0, stream>>>(h, ealpha, src, dst,
                                                         acc, NEDGES);
    elu_inplace<<<(NNODES * DHID + 255) / 256, 256, 0, stream>>>(acc, NNODES * DHID);

    // ---- conv4: h4 = h3 @ W4   [50000,512]@[512,1024] -> d_out+N*30 ----
    {   const int waves = superM * (DIN / 64);
        wmma_gemm_f32<4><<<(waves + 7) / 8, 256, 0, stream>>>(acc, W4,
                                                              out + (size_t)NNODES * DOUT,
                                                              NNODES, DIN, DHID);
    }
}